// CoAttention_23132693856407
// MI455X (gfx1250) — compile-verified
//
#include <hip/hip_runtime.h>
#include <hip/hip_bf16.h>

// ---------------- problem constants ----------------
constexpr int Lc  = 512;   // L1 == L2
constexpr int Bc  = 2;
constexpr int Dc  = 256;   // D1 == D2
constexpr int BNc = 256;
constexpr float NEGV = -1000000000000.0f;

// ---------------- workspace layout (floats) ---------
// P1: [b][l][k] (bh folded in), P2: [b][m][k]
constexpr size_t PSZ     = (size_t)Bc * Lc * BNc;        // 262144
constexpr size_t P1_OFF  = 0;
constexpr size_t P2_OFF  = PSZ;
constexpr size_t AFF_OFF = 2 * PSZ;                      // [b][l][m]
constexpr size_t ASZ     = (size_t)Bc * Lc * Lc;         // 524288
constexpr size_t D12_OFF = AFF_OFF + ASZ;
constexpr size_t D21_OFF = D12_OFF + ASZ;
constexpr size_t C12_OFF = D21_OFF + ASZ;                // [b][m][d]
constexpr size_t C21_OFF = C12_OFF + PSZ;                // [b][l][d]
// total = 2621440 floats = 10 MB

typedef __attribute__((ext_vector_type(2))) float v2f;
typedef __attribute__((ext_vector_type(8))) float v8f;
typedef __attribute__((ext_vector_type(4))) int   v4i;

// hardware tanh (CDNA5 TRANS op)
__device__ __forceinline__ float htanh(float x) {
#if __has_builtin(__builtin_amdgcn_tanhf)
  return __builtin_amdgcn_tanhf(x);
#else
  float r;
  asm("v_tanh_f32 %0, %1\n\tv_nop" : "=v"(r) : "v"(x));
  return r;
#endif
}

// CDNA5 async global->LDS copy (ASYNCcnt-tracked), with scalar fallback.
// Builtin signature (from hipcc diagnostic): param1 = v4i in AS1 (global),
// param2 = LDS side, then imm offset + imm cpol.
#if __has_builtin(__builtin_amdgcn_global_load_async_to_lds_b128) && \
    __has_builtin(__builtin_amdgcn_s_wait_asynccnt)
#define ASYNC_OK 1
typedef __attribute__((address_space(1))) v4i gas_v4i;
typedef __attribute__((address_space(3))) v4i las_v4i;
__device__ __forceinline__ void async_b128(const float* g, float* l) {
  __builtin_amdgcn_global_load_async_to_lds_b128(
      (gas_v4i*)(float*)g, (las_v4i*)l, 0, 0);
}
#endif

// ------------------------------------------------------------------
// Wave-level 16x16 fp32 WMMA tile: D(16x16) = sum_k A(16xK) * B(Kx16)
// A fragment: lane = M + 16*(k_half), vgpr j -> K = 2*(lane/16)+j
// B fragment: lane = N + 16*(k_half), vgpr j -> K = 2*(lane/16)+j
// C/D: vgpr r -> M = r + 8*(lane/16), N = lane%16
// loadA(r_rel, k_abs), loadB(k_abs, c_rel)
// ------------------------------------------------------------------
template <typename FA, typename FB>
__device__ __forceinline__ v8f wmma_tile_f32(FA loadA, FB loadB, int K) {
  const int lane = threadIdx.x & 31;
  const int half = lane >> 4;
  const int lr   = lane & 15;
  v8f acc = {};
  for (int k0 = 0; k0 < K; k0 += 4) {
    const int ka = k0 + 2 * half;
    v2f a, b;
    a[0] = loadA(lr, ka);
    a[1] = loadA(lr, ka + 1);
    b[0] = loadB(ka, lr);
    b[1] = loadB(ka + 1, lr);
    acc = __builtin_amdgcn_wmma_f32_16x16x4_f32(false, a, false, b,
                                                (short)0, acc, false, false);
  }
  return acc;
}

// ------------------------------------------------------------------
// Kernel 1: projections  p1[b][l][k] = ctx1[l,b,:]@Wh[:256] + bh
//                        p2[b][m][k] = ctx2[m,b,:]@Wh[256:]
// grid (32, 4, 4): x=row tile(16), y=col tile(64), z = sel*2+b. block 128.
// ------------------------------------------------------------------
__global__ void proj_kernel(const float* __restrict__ ctx1,
                            const float* __restrict__ ctx2,
                            const float* __restrict__ Wh,
                            const float* __restrict__ bh,
                            float* __restrict__ ws) {
  const int sel  = blockIdx.z >> 1;
  const int b    = blockIdx.z & 1;
  const int wave = threadIdx.x >> 5;
  const int row0 = blockIdx.x * 16;
  const int col0 = blockIdx.y * 64 + wave * 16;

  const float* ctx = sel ? ctx2 : ctx1;
  const float* W   = Wh + (size_t)sel * Dc * BNc;

  auto loadA = [&](int r, int k) {
    return ctx[((size_t)(row0 + r) * Bc + b) * Dc + k];
  };
  auto loadB = [&](int k, int c) {
    return W[(size_t)k * BNc + (col0 + c)];
  };
  v8f acc = wmma_tile_f32(loadA, loadB, Dc);

  float* P = ws + (sel ? P2_OFF : P1_OFF);
  const int lane = threadIdx.x & 31;
  const int half = lane >> 4, lr = lane & 15;
  const int c = col0 + lr;
  const float bias = sel ? 0.0f : bh[c];
#pragma unroll
  for (int r = 0; r < 8; ++r) {
    const int row = row0 + r + 8 * half;
    P[((size_t)b * Lc + row) * BNc + c] = acc[r] + bias;
  }
}

// ------------------------------------------------------------------
// Kernel 2: affinity[b][l][m] = sum_k tanh(p1'[l,k]+p2[m,k]) * wo[k] + masks
// 32x32 (l,m) tile per block, K staged in chunks of 128 through LDS
// (async global->LDS when available). TRANS(tanh)-bound by design.
// ------------------------------------------------------------------
constexpr int KCH  = 128;
constexpr int PADK = 132;   // 132 % 64banks = 4 -> conflict-free row stagger

__global__ void affinity_kernel(const float* __restrict__ m1,
                                const float* __restrict__ m2,
                                const float* __restrict__ wo,
                                float* __restrict__ ws) {
  const float* P1 = ws + P1_OFF;
  const float* P2 = ws + P2_OFF;
  float* AFF = ws + AFF_OFF;

  __shared__ float s1[32 * PADK];
  __shared__ float s2[32 * PADK];
  __shared__ float swo[BNc];

  const int b  = blockIdx.z;
  const int l0 = blockIdx.x * 32;
  const int m0 = blockIdx.y * 32;
  const int t  = threadIdx.x;

  swo[t] = wo[t];

  const int l  = t >> 3;   // 0..31
  const int mg = t & 7;    // 0..7

  float acc0 = 0.f, acc1 = 0.f, acc2 = 0.f, acc3 = 0.f;

  for (int kk = 0; kk < BNc; kk += KCH) {
    __syncthreads();   // protect LDS reuse across chunks (and swo on chunk 0)
#if defined(ASYNC_OK)
#pragma unroll
    for (int i = 0; i < 4; ++i) {
      const int f  = i * 256 + t;   // 1024 x b128 transfers per tile
      const int r  = f >> 5;        // 0..31
      const int c4 = (f & 31) * 4;  // 0..124, 16B-aligned both sides
      async_b128(&P1[((size_t)b * Lc + l0 + r) * BNc + kk + c4],
                 &s1[r * PADK + c4]);
      async_b128(&P2[((size_t)b * Lc + m0 + r) * BNc + kk + c4],
                 &s2[r * PADK + c4]);
    }
    __builtin_amdgcn_s_wait_asynccnt(0);
#else
#pragma unroll
    for (int i = 0; i < 16; ++i) {
      const int f = i * 256 + t;
      const int r = f >> 7;         // 0..31
      const int c = f & (KCH - 1);  // 0..127
      s1[r * PADK + c] = P1[((size_t)b * Lc + l0 + r) * BNc + kk + c];
      s2[r * PADK + c] = P2[((size_t)b * Lc + m0 + r) * BNc + kk + c];
    }
#endif
    __syncthreads();

    const float* r1 = &s1[l * PADK];
    const float* q0 = &s2[(mg +  0) * PADK];
    const float* q1 = &s2[(mg +  8) * PADK];
    const float* q2 = &s2[(mg + 16) * PADK];
    const float* q3 = &s2[(mg + 24) * PADK];
    const float* w  = &swo[kk];
#pragma unroll 8
    for (int k = 0; k < KCH; ++k) {
      const float a  = r1[k];
      const float wk = w[k];
      acc0 = fmaf(htanh(a + q0[k]), wk, acc0);
      acc1 = fmaf(htanh(a + q1[k]), wk, acc1);
      acc2 = fmaf(htanh(a + q2[k]), wk, acc2);
      acc3 = fmaf(htanh(a + q3[k]), wk, acc3);
    }
  }

  const int gl = l0 + l;
  const float addl = (1.0f - m1[(size_t)gl * Bc + b]) * NEGV;
  float* arow = AFF + ((size_t)b * Lc + gl) * Lc + m0;
  arow[mg +  0] = acc0 + addl + (1.0f - m2[(size_t)(m0 + mg +  0) * Bc + b]) * NEGV;
  arow[mg +  8] = acc1 + addl + (1.0f - m2[(size_t)(m0 + mg +  8) * Bc + b]) * NEGV;
  arow[mg + 16] = acc2 + addl + (1.0f - m2[(size_t)(m0 + mg + 16) * Bc + b]) * NEGV;
  arow[mg + 24] = acc3 + addl + (1.0f - m2[(size_t)(m0 + mg + 24) * Bc + b]) * NEGV;
}

// ------------------------------------------------------------------
// Kernel 3/4: softmax over m (rows) -> D12 ; over l (cols) -> D21
// grid (512, 2), block 256, each thread owns 2 of 512 elements.
// ------------------------------------------------------------------
__global__ void softmax_row_kernel(float* __restrict__ ws) {
  const float* AFF = ws + AFF_OFF;
  float* D12 = ws + D12_OFF;
  const int l = blockIdx.x, b = blockIdx.y, t = threadIdx.x;
  __shared__ float red[256];

  const float* row = AFF + ((size_t)b * Lc + l) * Lc;
  const float x0 = row[t], x1 = row[t + 256];
  red[t] = fmaxf(x0, x1);
  __syncthreads();
  for (int s = 128; s > 0; s >>= 1) {
    if (t < s) red[t] = fmaxf(red[t], red[t + s]);
    __syncthreads();
  }
  const float mx = red[0];
  __syncthreads();
  const float e0 = __expf(x0 - mx), e1 = __expf(x1 - mx);
  red[t] = e0 + e1;
  __syncthreads();
  for (int s = 128; s > 0; s >>= 1) {
    if (t < s) red[t] += red[t + s];
    __syncthreads();
  }
  const float inv = 1.0f / red[0];
  float* drow = D12 + ((size_t)b * Lc + l) * Lc;
  drow[t]       = e0 * inv;
  drow[t + 256] = e1 * inv;
}

__global__ void softmax_col_kernel(float* __restrict__ ws) {
  const float* AFF = ws + AFF_OFF;
  float* D21 = ws + D21_OFF;
  const int m = blockIdx.x, b = blockIdx.y, t = threadIdx.x;
  __shared__ float red[256];

  const float x0 = AFF[((size_t)b * Lc + t      ) * Lc + m];
  const float x1 = AFF[((size_t)b * Lc + t + 256) * Lc + m];
  red[t] = fmaxf(x0, x1);
  __syncthreads();
  for (int s = 128; s > 0; s >>= 1) {
    if (t < s) red[t] = fmaxf(red[t], red[t + s]);
    __syncthreads();
  }
  const float mx = red[0];
  __syncthreads();
  const float e0 = __expf(x0 - mx), e1 = __expf(x1 - mx);
  red[t] = e0 + e1;
  __syncthreads();
  for (int s = 128; s > 0; s >>= 1) {
    if (t < s) red[t] += red[t + s];
    __syncthreads();
  }
  const float inv = 1.0f / red[0];
  D21[((size_t)b * Lc + t      ) * Lc + m] = e0 * inv;
  D21[((size_t)b * Lc + t + 256) * Lc + m] = e1 * inv;
}

// ------------------------------------------------------------------
// Kernel 5: context GEMMs (K = 512)
//  sel==0: C12[b][m][d] = sum_l D12[b][l][m] * ctx1[l,b,d]   (A transposed)
//  sel==1: C21[b][l][d] = sum_m D21[b][l][m] * ctx2[m,b,d]
// grid (32, 4, 4), block 128.
// ------------------------------------------------------------------
__global__ void ctx_gemm_kernel(const float* __restrict__ ctx1,
                                const float* __restrict__ ctx2,
                                float* __restrict__ ws) {
  const int sel  = blockIdx.z >> 1;
  const int b    = blockIdx.z & 1;
  const int wave = threadIdx.x >> 5;
  const int row0 = blockIdx.x * 16;
  const int col0 = blockIdx.y * 64 + wave * 16;

  const float* D12 = ws + D12_OFF;
  const float* D21 = ws + D21_OFF;
  const float* ctx = sel ? ctx2 : ctx1;

  auto loadA = [&](int r, int k) {
    return sel == 0 ? D12[((size_t)b * Lc + k) * Lc + (row0 + r)]
                    : D21[((size_t)b * Lc + (row0 + r)) * Lc + k];
  };
  auto loadB = [&](int k, int c) {
    return ctx[((size_t)k * Bc + b) * Dc + (col0 + c)];
  };
  v8f acc = wmma_tile_f32(loadA, loadB, Lc);

  float* C = ws + (sel ? C21_OFF : C12_OFF);
  const int lane = threadIdx.x & 31;
  const int half = lane >> 4, lr = lane & 15;
  const int c = col0 + lr;
#pragma unroll
  for (int r = 0; r < 8; ++r) {
    const int row = row0 + r + 8 * half;
    C[((size_t)b * Lc + row) * Dc + c] = acc[r];
  }
}

// ------------------------------------------------------------------
// Kernel 6: output GEMMs (K = 512, concat done by address math) + tanh
//  sel==0: seq_2_to_1[l,b,:] = tanh([ctx1 | C21] @ W21 + b21) -> out[0..]
//  sel==1: seq_1_to_2[m,b,:] = tanh([ctx2 | C12] @ W12 + b12) -> out[L*B*BN..]
// grid (32, 4, 4), block 128.
// ------------------------------------------------------------------
__global__ void out_gemm_kernel(const float* __restrict__ ctx1,
                                const float* __restrict__ ctx2,
                                const float* __restrict__ W12,
                                const float* __restrict__ b12,
                                const float* __restrict__ W21,
                                const float* __restrict__ b21,
                                const float* __restrict__ ws,
                                float* __restrict__ out) {
  const int sel  = blockIdx.z >> 1;
  const int b    = blockIdx.z & 1;
  const int wave = threadIdx.x >> 5;
  const int row0 = blockIdx.x * 16;
  const int col0 = blockIdx.y * 64 + wave * 16;

  const float* Xc = sel ? ctx2 : ctx1;
  const float* Cc = ws + (sel ? C12_OFF : C21_OFF);
  const float* W  = sel ? W12 : W21;
  const float* bi = sel ? b12 : b21;
  const size_t obase = sel ? (size_t)Lc * Bc * BNc : 0;

  auto loadA = [&](int r, int j) {
    return j < Dc ? Xc[((size_t)(row0 + r) * Bc + b) * Dc + j]
                  : Cc[((size_t)b * Lc + (row0 + r)) * Dc + (j - Dc)];
  };
  auto loadB = [&](int j, int c) {
    return W[(size_t)j * BNc + (col0 + c)];
  };
  v8f acc = wmma_tile_f32(loadA, loadB, 2 * Dc);

  const int lane = threadIdx.x & 31;
  const int half = lane >> 4, lr = lane & 15;
  const int c = col0 + lr;
  const float bias = bi[c];
#pragma unroll
  for (int r = 0; r < 8; ++r) {
    const int row = row0 + r + 8 * half;
    out[obase + ((size_t)row * Bc + b) * BNc + c] = htanh(acc[r] + bias);
  }
}

// ------------------------------------------------------------------
extern "C" void kernel_launch(void* const* d_in, const int* in_sizes, int n_in,
                              void* d_out, int out_size, void* d_ws, size_t ws_size,
                              hipStream_t stream) {
  const float* ctx1 = (const float*)d_in[0];
  const float* ctx2 = (const float*)d_in[1];
  const float* m1   = (const float*)d_in[2];
  const float* m2   = (const float*)d_in[3];
  const float* Wh   = (const float*)d_in[4];
  const float* bh   = (const float*)d_in[5];
  const float* wo   = (const float*)d_in[6];
  const float* W12  = (const float*)d_in[7];
  const float* b12  = (const float*)d_in[8];
  const float* W21  = (const float*)d_in[9];
  const float* b21  = (const float*)d_in[10];
  float* out = (float*)d_out;
  float* ws  = (float*)d_ws;   // uses 10 MB

  proj_kernel     <<<dim3(Lc/16, BNc/64, 2*Bc), 128, 0, stream>>>(ctx1, ctx2, Wh, bh, ws);
  affinity_kernel <<<dim3(Lc/32, Lc/32, Bc),    256, 0, stream>>>(m1, m2, wo, ws);
  softmax_row_kernel<<<dim3(Lc, Bc), 256, 0, stream>>>(ws);
  softmax_col_kernel<<<dim3(Lc, Bc), 256, 0, stream>>>(ws);
  ctx_gemm_kernel <<<dim3(Lc/16, Dc/64, 2*Bc),  128, 0, stream>>>(ctx1, ctx2, ws);
  out_gemm_kernel <<<dim3(Lc/16, BNc/64, 2*Bc), 128, 0, stream>>>(ctx1, ctx2, W12, b12, W21, b21, ws, out);
}